// YOLODecoder_57406532879075
// MI455X (gfx1250) — compile-verified
//
#include <hip/hip_runtime.h>
#include <hip/hip_bf16.h>
#include <cstdint>

// ---------------- problem constants ----------------
#define NBOX        327680      // 256*256*5
#define CDIM        85
#define NCLASS      80
#define GRIDW       256
#define NANCH       5
#define NBINS       4096
#define TOPK        1024
#define CAP         4096
#define SORTN       4096
#define MAXOUT      100
#define DEC_ROWS    128
#define DEC_THREADS 128
#define TILE_ELEMS  (DEC_ROWS * CDIM)    // 10,880 dwords = 43,520 B per block

__device__ __forceinline__ float sigmoidf_(float v) {
    return 1.0f / (1.0f + __expf(-v));
}

// ---------------- TDM descriptor types ----------------
typedef __attribute__((ext_vector_type(4))) unsigned int tdm_u32x4;
typedef __attribute__((ext_vector_type(8))) int          tdm_i32x8;
typedef __attribute__((ext_vector_type(4))) int          tdm_i32x4;

// One-shot TDM DMA: contiguous nElems dwords global -> LDS.
// Descriptor: 2D tensor, 1 row, tile_dim0 = nElems (fits 16 bits), data_size=4B.
__device__ __forceinline__ void tdm_copy_to_lds(unsigned ldsAddr,
                                                unsigned long long gaddr,
                                                unsigned nElems)
{
#if __has_builtin(__builtin_amdgcn_tensor_load_to_lds)
    tdm_u32x4 g0 = {
        1u,                                                   // count=1, user mode
        ldsAddr,                                              // lds_addr (bytes)
        (unsigned)(gaddr & 0xFFFFFFFFu),                      // global_addr[31:0]
        (unsigned)((gaddr >> 32) & 0x01FFFFFFu) | (2u << 30)  // global_addr[56:32] | type=2
    };
    tdm_i32x8 g1 = {
        (int)(2u << 16),                       // wg_mask=0, data_size=2 (4 bytes)
        (int)((nElems & 0xFFFFu) << 16),       // tensor_dim0[15:0]
        (int)(((nElems >> 16) & 0xFFFFu) | (1u << 16)), // tensor_dim0[31:16], tensor_dim1=1
        (int)((nElems & 0xFFFFu) << 16),       // tensor_dim1[31:16]=0, tile_dim0=nElems
        1,                                     // tile_dim1=1, tile_dim2=0
        (int)nElems,                           // tensor_dim0_stride[31:0]
        0, 0                                   // stride0[47:32]=0, stride1=0
    };
    tdm_i32x4 z = { 0, 0, 0, 0 };
#if defined(__clang_major__) && (__clang_major__ >= 23)
    tdm_i32x8 z8 = { 0, 0, 0, 0, 0, 0, 0, 0 };
    __builtin_amdgcn_tensor_load_to_lds(g0, g1, z, z, z8, 0);
#else
    __builtin_amdgcn_tensor_load_to_lds(g0, g1, z, z, 0);
#endif
#else
    (void)ldsAddr; (void)gaddr; (void)nElems;
#endif
}

// ---------------- kernel 0: zero hist + counters ----------------
__global__ void zero_kernel(unsigned* __restrict__ hist) {
    int i = blockIdx.x * blockDim.x + threadIdx.x;
    if (i < NBINS + 64) hist[i] = 0u;   // covers histogram, counter, thresh slot
}

// ---------------- kernel 1: decode (BW bound; TDM DMA staging) ----------------
__global__ __launch_bounds__(DEC_THREADS)
void decode_kernel(const float* __restrict__ x, const float* __restrict__ anchors,
                   float* __restrict__ scores,
                   float* __restrict__ bx1, float* __restrict__ by1,
                   float* __restrict__ bx2, float* __restrict__ by2,
                   int*   __restrict__ cls,
                   unsigned* __restrict__ hist)
{
    __shared__ float tile[TILE_ELEMS];                // 43,520 B

    const int t       = threadIdx.x;
    const int rowBase = blockIdx.x * DEC_ROWS;
    const float* src  = x + (size_t)rowBase * CDIM;

    // Raw LDS byte offset (generic-address low 32 bits == wave-relative LDS offset)
    unsigned ldsBase = (unsigned)(uintptr_t)(&tile[0]);
    unsigned long long srcAddr = (unsigned long long)(uintptr_t)src;

#if __has_builtin(__builtin_amdgcn_tensor_load_to_lds)
    // One TDM descriptor moves the whole contiguous tile; wave 0 issues it,
    // waits on TENSORcnt, then the workgroup barrier publishes the LDS tile.
    if (t < 32) {
        tdm_copy_to_lds(ldsBase, srcAddr, (unsigned)TILE_ELEMS);
        __builtin_amdgcn_s_wait_tensorcnt(0);
    }
    __syncthreads();
#else
    // Fallback (round-1 path): coalesced async global->LDS b32 staging.
    #pragma unroll 5
    for (int k = 0; k < CDIM; ++k) {
        unsigned off = (unsigned)((k * DEC_THREADS + t) * 4);
        unsigned lds = ldsBase + off;
        asm volatile("global_load_async_to_lds_b32 %0, %1, %2"
                     :: "v"(lds), "v"(off), "s"(srcAddr)
                     : "memory");
    }
    asm volatile("s_wait_asynccnt 0" ::: "memory");
    __syncthreads();
#endif

    const float* my = &tile[t * CDIM];   // stride 85 words: conflict-free banks

    // softmax max / argmax over 80 class logits
    float lmax = -1e30f; int amax = 0;
    #pragma unroll 8
    for (int j = 0; j < NCLASS; ++j) {
        float l = my[5 + j];
        if (l > lmax) { lmax = l; amax = j; }
    }
    float ssum = 0.f;
    #pragma unroll 8
    for (int j = 0; j < NCLASS; ++j) ssum += __expf(my[5 + j] - lmax);
    float pmax = 1.0f / ssum;                 // max of softmax
    float conf = sigmoidf_(my[4]);
    float best = conf * pmax;                 // max class confidence

    float score; int c;
    if (best > 0.01f) { score = best; c = amax; }
    else              { score = 0.f;  c = 0;    }   // masked-out: argmax of zeros == 0

    int n      = rowBase + t;
    int a      = n % NANCH;
    int cellid = n / NANCH;
    int cw     = cellid & (GRIDW - 1);
    int ch     = cellid >> 8;

    float gx = (sigmoidf_(my[0]) + (float)cw) * (1.0f / (float)GRIDW);
    float gy = (sigmoidf_(my[1]) + (float)ch) * (1.0f / (float)GRIDW);
    float bw = __expf(my[2]) * anchors[a * 2 + 0] * (1.0f / (float)GRIDW);
    float bh = __expf(my[3]) * anchors[a * 2 + 1] * (1.0f / (float)GRIDW);

    bx1[n] = gx - 0.5f * bw;  by1[n] = gy - 0.5f * bh;
    bx2[n] = gx + 0.5f * bw;  by2[n] = gy + 0.5f * bh;
    scores[n] = score;
    cls[n]    = c;

    if (score > 0.f) {
        int b = (int)(score * (float)NBINS);
        if (b > NBINS - 1) b = NBINS - 1;
        atomicAdd(&hist[b], 1u);
    }
}

// ---------------- kernel 2: pick score threshold covering >= TOPK ----------------
__global__ void thresh_kernel(const unsigned* __restrict__ hist, float* __restrict__ thresh)
{
    if (threadIdx.x == 0) {
        unsigned cum = 0; int sel = -1;
        for (int b = NBINS - 1; b >= 0; --b) {
            cum += hist[b];
            if (cum >= TOPK) { sel = b; break; }
        }
        *thresh = (sel >= 0) ? ((float)sel / (float)NBINS) : 0.0f;
    }
}

// ---------------- kernel 3: compact candidates ----------------
__global__ void compact_kernel(const float* __restrict__ scores,
                               const float* __restrict__ thresh,
                               unsigned* __restrict__ counter,
                               float* __restrict__ cand_s, int* __restrict__ cand_i)
{
    int n = blockIdx.x * blockDim.x + threadIdx.x;
    if (n >= NBOX) return;
    float s = scores[n];
    float th = *thresh;
    if (s > 0.f && s >= th) {
        unsigned pos = atomicAdd(counter, 1u);
        if (pos < CAP) { cand_s[pos] = s; cand_i[pos] = n; }
    }
}

// ---------------- kernel 4: single-block bitonic sort (desc score, asc idx) ----------------
__global__ __launch_bounds__(1024)
void sort_kernel(const float* __restrict__ cand_s, const int* __restrict__ cand_i,
                 const unsigned* __restrict__ counter,
                 const float* __restrict__ bx1, const float* __restrict__ by1,
                 const float* __restrict__ bx2, const float* __restrict__ by2,
                 const int* __restrict__ cls,
                 float* __restrict__ s_sc, int* __restrict__ s_idx,
                 float* __restrict__ sbx1, float* __restrict__ sby1,
                 float* __restrict__ sbx2, float* __restrict__ sby2,
                 int* __restrict__ s_cls)
{
    __shared__ float sc[SORTN];
    __shared__ int   id[SORTN];

    unsigned cc = *counter;
    int count = (int)(cc < (unsigned)CAP ? cc : (unsigned)CAP);
    int tid = threadIdx.x;

    for (int e = tid; e < SORTN; e += 1024) {
        if (e < count) { sc[e] = cand_s[e]; id[e] = cand_i[e]; }
        else           { sc[e] = -1.0f;     id[e] = 0x7fffffff; }
    }
    __syncthreads();

    for (int k = 2; k <= SORTN; k <<= 1) {
        for (int j = k >> 1; j > 0; j >>= 1) {
            for (int e = tid; e < SORTN; e += 1024) {
                int p = e ^ j;
                if (p > e) {
                    float se = sc[e], sp = sc[p];
                    int   ie = id[e], ip = id[p];
                    // "precedes": higher score first, ties -> lower original index
                    bool pPrec = (sp > se) || (sp == se && ip < ie);
                    bool fwd   = ((e & k) == 0);
                    bool doSwap = fwd ? pPrec : !pPrec;
                    if (doSwap) { sc[e] = sp; sc[p] = se; id[e] = ip; id[p] = ie; }
                }
            }
            __syncthreads();
        }
    }

    for (int e = tid; e < SORTN; e += 1024) {
        s_sc[e] = sc[e]; s_idx[e] = id[e];
        if (e < count) {
            int g = id[e];
            sbx1[e] = bx1[g]; sby1[e] = by1[g];
            sbx2[e] = bx2[g]; sby2[e] = by2[g];
            s_cls[e] = cls[g];
        }
    }
}

// ---------------- kernel 5: suppression bit-matrix ----------------
__global__ __launch_bounds__(64)
void iou_kernel(const float* __restrict__ sbx1, const float* __restrict__ sby1,
                const float* __restrict__ sbx2, const float* __restrict__ sby2,
                const unsigned* __restrict__ counter,
                unsigned long long* __restrict__ mat)
{
    int i = blockIdx.x;
    unsigned cc = *counter;
    int count = (int)(cc < (unsigned)CAP ? cc : (unsigned)CAP);
    if (i >= count) return;

    int t = threadIdx.x;                 // thread t owns 64-bit word t of row i
    float x1i = sbx1[i], y1i = sby1[i], x2i = sbx2[i], y2i = sby2[i];
    float ai  = (x2i - x1i) * (y2i - y1i);

    unsigned long long bits = 0ull;
    int jbase = t * 64;
    #pragma unroll 4
    for (int b = 0; b < 64; ++b) {
        int j = jbase + b;
        if (j > i && j < count) {
            float xx1 = fmaxf(x1i, sbx1[j]);
            float yy1 = fmaxf(y1i, sby1[j]);
            float xx2 = fminf(x2i, sbx2[j]);
            float yy2 = fminf(y2i, sby2[j]);
            float iw  = fmaxf(xx2 - xx1, 0.f);
            float ih  = fmaxf(yy2 - yy1, 0.f);
            float inter = iw * ih;
            float aj  = (sbx2[j] - sbx1[j]) * (sby2[j] - sby1[j]);
            float uni = ai + aj - inter;
            if (uni > 0.f && inter > 0.5f * uni)   // iou > 0.5
                bits |= (1ull << b);
        }
    }
    mat[(size_t)i * (SORTN / 64) + t] = bits;
}

// ---------------- kernel 6: single-wave greedy scan + output ----------------
__global__ __launch_bounds__(32)
void nms_scan_kernel(const float* __restrict__ s_sc,
                     const float* __restrict__ sbx1, const float* __restrict__ sby1,
                     const float* __restrict__ sbx2, const float* __restrict__ sby2,
                     const int* __restrict__ s_cls,
                     const unsigned* __restrict__ counter,
                     const unsigned long long* __restrict__ mat,
                     float* __restrict__ out)
{
    int lane = threadIdx.x;
    unsigned cc = *counter;
    int count = (int)(cc < (unsigned)CAP ? cc : (unsigned)CAP);

    // 4096-bit removal bitmap: 64 words, lane owns words 2*lane, 2*lane+1
    unsigned long long r0 = 0ull, r1 = 0ull;
    int kept = 0;

    for (int i = 0; i < count && kept < MAXOUT; ++i) {
        float sc = s_sc[i];
        if (sc <= 0.f) break;                       // no more valid picks
        int w     = i >> 6;                         // word index (uniform)
        int owner = w >> 1;
        unsigned long long v = (w & 1) ? r1 : r0;   // (w&1) uniform across wave
        v = __shfl(v, owner, 32);
        if (!((v >> (i & 63)) & 1ull)) {
            // keep i: merge its suppression row
            const unsigned long long* row = mat + (size_t)i * (SORTN / 64);
            r0 |= row[2 * lane + 0];
            r1 |= row[2 * lane + 1];
            if (lane == 0) {
                out[kept]                  = sc;
                out[MAXOUT + kept * 4 + 0] = sbx1[i];
                out[MAXOUT + kept * 4 + 1] = sby1[i];
                out[MAXOUT + kept * 4 + 2] = sbx2[i];
                out[MAXOUT + kept * 4 + 3] = sby2[i];
                out[MAXOUT * 5 + kept]     = (float)s_cls[i];
            }
            kept++;
        }
    }

    // pad remaining slots exactly like reference invalid picks
    for (int j = kept + lane; j < MAXOUT; j += 32) {
        out[j] = 0.f;
        out[MAXOUT + j * 4 + 0] = 0.f;
        out[MAXOUT + j * 4 + 1] = 0.f;
        out[MAXOUT + j * 4 + 2] = 0.f;
        out[MAXOUT + j * 4 + 3] = 0.f;
        out[MAXOUT * 5 + j] = -1.0f;
    }
}

// ---------------- host-side launcher ----------------
extern "C" void kernel_launch(void* const* d_in, const int* in_sizes, int n_in,
                              void* d_out, int out_size, void* d_ws, size_t ws_size,
                              hipStream_t stream)
{
    (void)in_sizes; (void)n_in; (void)out_size; (void)ws_size;

    const float* x       = (const float*)d_in[0];
    const float* anchors = (const float*)d_in[1];
    float* out = (float*)d_out;
    char*  ws  = (char*)d_ws;

    const size_t NB = (size_t)NBOX * 4;             // 1,310,720 B per f32 array

    float*    scores  = (float*)(ws + 0 * NB);
    float*    bx1     = (float*)(ws + 1 * NB);
    float*    by1     = (float*)(ws + 2 * NB);
    float*    bx2     = (float*)(ws + 3 * NB);
    float*    by2     = (float*)(ws + 4 * NB);
    int*      cls     = (int*)  (ws + 5 * NB);
    unsigned* hist    = (unsigned*)(ws + 6 * NB);           // NBINS u32
    unsigned* counter = hist + NBINS;                       // [0] = cand count
    float*    thresh  = (float*)(counter + 1);

    char* p = ws + 6 * NB + NBINS * 4 + 256;                // 256B-aligned tail
    float* cand_s = (float*)(p + 0 * (CAP * 4));
    int*   cand_i = (int*)  (p + 1 * (CAP * 4));
    float* s_sc   = (float*)(p + 2 * (CAP * 4));
    int*   s_idx  = (int*)  (p + 3 * (CAP * 4));
    float* sbx1   = (float*)(p + 4 * (CAP * 4));
    float* sby1   = (float*)(p + 5 * (CAP * 4));
    float* sbx2   = (float*)(p + 6 * (CAP * 4));
    float* sby2   = (float*)(p + 7 * (CAP * 4));
    int*   s_cls  = (int*)  (p + 8 * (CAP * 4));
    unsigned long long* mat = (unsigned long long*)(p + 9 * (CAP * 4)); // 2 MB

    zero_kernel<<<(NBINS + 64 + 255) / 256, 256, 0, stream>>>(hist);

    decode_kernel<<<NBOX / DEC_ROWS, DEC_THREADS, 0, stream>>>(
        x, anchors, scores, bx1, by1, bx2, by2, cls, hist);

    thresh_kernel<<<1, 32, 0, stream>>>(hist, thresh);

    compact_kernel<<<(NBOX + 255) / 256, 256, 0, stream>>>(
        scores, thresh, counter, cand_s, cand_i);

    sort_kernel<<<1, 1024, 0, stream>>>(
        cand_s, cand_i, counter, bx1, by1, bx2, by2, cls,
        s_sc, s_idx, sbx1, sby1, sbx2, sby2, s_cls);

    iou_kernel<<<CAP, 64, 0, stream>>>(sbx1, sby1, sbx2, sby2, counter, mat);

    nms_scan_kernel<<<1, 32, 0, stream>>>(
        s_sc, sbx1, sby1, sbx2, sby2, s_cls, counter, mat, out);
}